// GNNModel_76785425318467
// MI455X (gfx1250) — compile-verified
//
#include <hip/hip_runtime.h>

#define NNODES 100000
#define NEDGES 1200000
#define IN_DIM 64
#define HID_DIM 64
#define OUT_DIM 32

typedef __attribute__((ext_vector_type(2))) float v2f;
typedef __attribute__((ext_vector_type(4))) float v4f;
typedef __attribute__((ext_vector_type(8))) float v8f;

// ---------------------------------------------------------------------------
// Utility: fill a float buffer with a constant
// ---------------------------------------------------------------------------
__global__ void gcn_fill_kernel(float* __restrict__ p, float v, int n) {
    int i = blockIdx.x * blockDim.x + threadIdx.x;
    if (i < n) p[i] = v;
}

// ---------------------------------------------------------------------------
// Degree count: deg[dst[e]] += 1  (deg pre-filled with 1.0 for self-loops)
// ---------------------------------------------------------------------------
__global__ void gcn_deg_kernel(const int* __restrict__ dst, float* __restrict__ deg, int E) {
    int e = blockIdx.x * blockDim.x + threadIdx.x;
    if (e < E) atomicAdd(&deg[dst[e]], 1.0f);
}

// In-place deg -> deg^{-1/2}
__global__ void gcn_rsqrt_kernel(float* __restrict__ deg, int n) {
    int i = blockIdx.x * blockDim.x + threadIdx.x;
    if (i < n) deg[i] = rsqrtf(deg[i]);
}

// ---------------------------------------------------------------------------
// WMMA f32 GEMM with A-fragment reuse: C[M,N] = A[M,K] * B[K,N], row-major.
// One wave32 per 16-row slab; the wave owns ALL N/16 column tiles so each A
// fragment is loaded once and feeds N/16 V_WMMA_F32_16X16X4_F32 ops.
//
// A 16x4 layout : lane = (m=lane&15, khalf=lane>>4); VGPR j -> K = khalf*2+j
//                 -> the two A elements are consecutive in memory (1x b64 load)
// B 4x16 layout : lane holds column n=lane&15; VGPR j -> K = khalf*2+j
// C/D 16x16     : VGPR j, lane half -> row = j + 8*(lane>=16), col = lane&15
// ---------------------------------------------------------------------------
template <int K, int N>
__global__ __launch_bounds__(256) void gcn_gemm_wmma_kernel(
    const float* __restrict__ A, const float* __restrict__ B,
    float* __restrict__ C, int M) {
    constexpr int NT = N / 16;  // column tiles per wave (4 or 2)

    const int lane     = threadIdx.x & 31;
    const int wave     = threadIdx.x >> 5;
    const int row_tile = blockIdx.x * 8 + wave;
    if (row_tile * 16 >= M) return;

    const int m     = lane & 15;
    const int khalf = lane >> 4;          // 0 or 1
    const int row   = row_tile * 16 + m;  // A row this lane feeds

    v8f acc[NT];
#pragma unroll
    for (int nt = 0; nt < NT; ++nt) acc[nt] = (v8f){0.f,0.f,0.f,0.f,0.f,0.f,0.f,0.f};

    const float* __restrict__ arow = A + (size_t)row * K;
#pragma unroll
    for (int k0 = 0; k0 < K; k0 += 4) {
        const int kk = k0 + khalf * 2;
        v2f a = *(const v2f*)(arow + kk);            // A[row][kk], A[row][kk+1]
#pragma unroll
        for (int nt = 0; nt < NT; ++nt) {
            const int n = nt * 16 + m;               // B column this lane feeds
            v2f b;
            b.x = B[(size_t)kk * N + n];             // B[kk][n]
            b.y = B[(size_t)(kk + 1) * N + n];       // B[kk+1][n]
            // 8 args: (neg_a, A, neg_b, B, c_mod, C, reuse_a, reuse_b)
            acc[nt] = __builtin_amdgcn_wmma_f32_16x16x4_f32(
                false, a, false, b, (short)0, acc[nt], false, false);
        }
    }

    // Store: lane (half) owns rows row_tile*16 + khalf*8 + j.
#pragma unroll
    for (int nt = 0; nt < NT; ++nt) {
        float* __restrict__ crow =
            C + (size_t)(row_tile * 16 + khalf * 8) * N + nt * 16 + m;
#pragma unroll
        for (int j = 0; j < 8; ++j) crow[(size_t)j * N] = acc[nt][j];
    }
}

// ---------------------------------------------------------------------------
// Edge scatter: out[dst[e], f] += h[src[e], f] * isq[src]*isq[dst]
// One thread per (edge, 4 features): one b128 gather + 4 f32 atomics.
// F/4 consecutive threads share an edge -> contiguous gather, broadcast index
// loads. Atomic-add volume (the real limiter) is unchanged.
// ---------------------------------------------------------------------------
template <int F>
__global__ __launch_bounds__(256) void gcn_scatter_kernel(
    const float* __restrict__ h, const int* __restrict__ src,
    const int* __restrict__ dst, const float* __restrict__ isq,
    float* __restrict__ out, int E) {
    constexpr int Q = F / 4;                 // float4 chunks per edge
    unsigned t = blockIdx.x * blockDim.x + threadIdx.x;
    unsigned e = t / Q;
    unsigned q = t % Q;
    if (e >= (unsigned)E) return;
    int s = src[e];
    int d = dst[e];
    float nrm = isq[s] * isq[d];
    v4f msg = *(const v4f*)(h + (unsigned)s * F + q * 4);
    float* o = out + (unsigned)d * F + q * 4;
    atomicAdd(o + 0, msg.x * nrm);
    atomicAdd(o + 1, msg.y * nrm);
    atomicAdd(o + 2, msg.z * nrm);
    atomicAdd(o + 3, msg.w * nrm);
}

// ---------------------------------------------------------------------------
// Fused epilogue: out += self-loop message (h[i]*isq[i]^2) + bias, optional ReLU
// ---------------------------------------------------------------------------
template <int F, bool RELU>
__global__ __launch_bounds__(256) void gcn_finish_kernel(
    const float* __restrict__ h, const float* __restrict__ isq,
    const float* __restrict__ bias, float* __restrict__ out, int n) {
    unsigned idx = blockIdx.x * blockDim.x + threadIdx.x;
    if (idx >= (unsigned)n * F) return;
    unsigned i = idx / F;
    unsigned f = idx % F;
    float si = isq[i];
    float v  = out[idx] + h[idx] * si * si + bias[f];
    out[idx] = RELU ? fmaxf(v, 0.0f) : v;
}

// ---------------------------------------------------------------------------
// Host-side orchestration (all launches on `stream`, graph-capture safe)
// ---------------------------------------------------------------------------
extern "C" void kernel_launch(void* const* d_in, const int* in_sizes, int n_in,
                              void* d_out, int out_size, void* d_ws, size_t ws_size,
                              hipStream_t stream) {
    const float* x  = (const float*)d_in[0];   // [NNODES, 64]
    const int*   ei = (const int*)d_in[1];     // [2, NEDGES]
    const float* W1 = (const float*)d_in[2];   // [64, 64]
    const float* b1 = (const float*)d_in[3];   // [64]
    const float* W2 = (const float*)d_in[4];   // [64, 32]
    const float* b2 = (const float*)d_in[5];   // [32]
    const int* src = ei;
    const int* dst = ei + NEDGES;
    float* out = (float*)d_out;                // [NNODES, 32]

    // Workspace layout (floats): isq | h1 (reused as h2) | agg1
    float* ws   = (float*)d_ws;
    float* isq  = ws;                                   // NNODES
    float* h1   = ws + 131072;                          // NNODES*64
    float* agg1 = h1 + (size_t)NNODES * HID_DIM;        // NNODES*64
    float* h2   = h1;                                   // NNODES*32 (reuse)

    const int T = 256;

    // --- degree + normalization --------------------------------------------
    gcn_fill_kernel<<<(NNODES + T - 1) / T, T, 0, stream>>>(isq, 1.0f, NNODES);
    gcn_deg_kernel<<<(NEDGES + T - 1) / T, T, 0, stream>>>(dst, isq, NEDGES);
    gcn_rsqrt_kernel<<<(NNODES + T - 1) / T, T, 0, stream>>>(isq, NNODES);

    // --- layer 1: h1 = x @ W1 (WMMA), aggregate, +b1, ReLU -----------------
    {
        int waves  = NNODES / 16;       // one wave per 16-row slab (all col tiles)
        int blocks = (waves + 7) / 8;
        gcn_gemm_wmma_kernel<IN_DIM, HID_DIM><<<blocks, T, 0, stream>>>(x, W1, h1, NNODES);
    }
    gcn_fill_kernel<<<((NNODES * HID_DIM) + T - 1) / T, T, 0, stream>>>(agg1, 0.0f, NNODES * HID_DIM);
    {
        unsigned total = (unsigned)NEDGES * (HID_DIM / 4);
        gcn_scatter_kernel<HID_DIM><<<(total + T - 1) / T, T, 0, stream>>>(h1, src, dst, isq, agg1, NEDGES);
    }
    gcn_finish_kernel<HID_DIM, true><<<((NNODES * HID_DIM) + T - 1) / T, T, 0, stream>>>(h1, isq, b1, agg1, NNODES);

    // --- layer 2: h2 = agg1 @ W2 (WMMA), aggregate into d_out, +b2 ---------
    {
        int waves  = NNODES / 16;
        int blocks = (waves + 7) / 8;
        gcn_gemm_wmma_kernel<HID_DIM, OUT_DIM><<<blocks, T, 0, stream>>>(agg1, W2, h2, NNODES);
    }
    gcn_fill_kernel<<<((NNODES * OUT_DIM) + T - 1) / T, T, 0, stream>>>(out, 0.0f, NNODES * OUT_DIM);
    {
        unsigned total = (unsigned)NEDGES * (OUT_DIM / 4);
        gcn_scatter_kernel<OUT_DIM><<<(total + T - 1) / T, T, 0, stream>>>(h2, src, dst, isq, out, NEDGES);
    }
    gcn_finish_kernel<OUT_DIM, false><<<((NNODES * OUT_DIM) + T - 1) / T, T, 0, stream>>>(h2, isq, b2, out, NNODES);
}